// OneLayer_RNN_88536455840383
// MI455X (gfx1250) — compile-verified
//
#include <hip/hip_runtime.h>
#include <cstdint>
#include <cstddef>

#define LSEQ 4096
#define BB   64
#define CC   256
#define HH   256

typedef float v2f __attribute__((ext_vector_type(2)));
typedef float v8f __attribute__((ext_vector_type(8)));

// ---------------------------------------------------------------------------
// Kernel 1: xW[l,b,:] = x[l,b,:] @ W_ih^T + bias
// One block = 16 rows of (L*B) x 256 output, 16 waves, one 16x16 N-tile each.
// A tile staged in LDS (padded stride 260 -> conflict-free K-column reads),
// B read from global (W_ih is 256KB, L2-resident). f32 WMMA 16x16x4,
// dual accumulators to break the C-chain.
// ---------------------------------------------------------------------------
__global__ __launch_bounds__(512) void rnn_xproj_wmma(
    const float* __restrict__ x, const float* __restrict__ W_ih,
    const float* __restrict__ bias, float* __restrict__ xw_out)
{
    __shared__ float As[16 * 260];
    const int tid  = threadIdx.x;
    const int wave = tid >> 5;
    const int lane = tid & 31;
    const int hi   = lane >> 4;
    const int col  = lane & 15;            // A-row for lane, and B/N column
    const int m0   = blockIdx.x * 16;      // global row base in (L*B)

    // Cooperative load of A tile: 16 rows x 256 K (coalesced)
    #pragma unroll
    for (int i = 0; i < 8; ++i) {
        int idx = tid + i * 512;
        int r = idx >> 8;
        int c = idx & 255;
        As[r * 260 + c] = x[(size_t)(m0 + r) * CC + c];
    }
    __syncthreads();

    const int n0 = wave * 16;
    const float2* Wp = reinterpret_cast<const float2*>(W_ih);
    const int brow = n0 + col;             // W_ih row feeding this lane's N column

    v8f acc0 = {};
    v8f acc1 = {};
    #pragma unroll
    for (int k0 = 0; k0 < CC; k0 += 8) {
        // A 16x4 f32 layout: lane holds A[lane&15][k + 2*hi + {0,1}]
        v2f a0 = *reinterpret_cast<const v2f*>(&As[col * 260 + k0 + 2 * hi]);
        v2f a1 = *reinterpret_cast<const v2f*>(&As[col * 260 + k0 + 4 + 2 * hi]);
        // B 4x16 layout: lane holds B[2*hi + v][col] = W_ih[n0+col][k + 2*hi + v]
        float2 bl0 = Wp[(size_t)brow * (CC / 2) + ((k0 + 2 * hi) >> 1)];
        float2 bl1 = Wp[(size_t)brow * (CC / 2) + ((k0 + 4 + 2 * hi) >> 1)];
        v2f b0; b0.x = bl0.x; b0.y = bl0.y;
        v2f b1; b1.x = bl1.x; b1.y = bl1.y;
        acc0 = __builtin_amdgcn_wmma_f32_16x16x4_f32(
            false, a0, false, b0, (short)0, acc0, false, false);
        acc1 = __builtin_amdgcn_wmma_f32_16x16x4_f32(
            false, a1, false, b1, (short)0, acc1, false, false);
    }
    v8f acc = acc0 + acc1;

    const float bv = bias[n0 + col];
    #pragma unroll
    for (int r = 0; r < 8; ++r) {
        int row = m0 + r + 8 * hi;         // C/D layout: M = vgpr + 8*hi, N = lane&15
        xw_out[(size_t)row * HH + n0 + col] = acc[r] + bv;
    }
}

// ---------------------------------------------------------------------------
// Kernel 2: sequential scan, batch-parallel across 4 independent workgroups
// (batch rows are independent in the recurrence -> no cross-WG sync ever).
// W_hh (256KB) + 16-row h slice live in LDS for all 4096 steps.
// Reads xW[l] from d_out and overwrites the same slot with h[l] in place.
// Dual accumulators per tile double WMMA ILP on the serial critical path.
// ---------------------------------------------------------------------------
__global__ __launch_bounds__(512) void rnn_scan_wmma(
    const float* __restrict__ W_hh, const float* __restrict__ h0,
    float* __restrict__ out, float* __restrict__ hlast)
{
    extern __shared__ float smem[];
    float* Wsh = smem;                     // [256][258] padded
    float* Hs  = smem + 256 * 258;         // [16][258]  padded

    const int tid  = threadIdx.x;
    const int wave = tid >> 5;
    const int lane = tid & 31;
    const int hi   = lane >> 4;
    const int col  = lane & 15;
    const int b0   = blockIdx.x * 16;      // this WG's batch-row slice

    // Stage W_hh (row-major [j][k], B-matrix is W_hh^T so b[k][n]=W_hh[n][k])
    for (int idx = tid; idx < HH * HH; idx += 512) {
        int j = idx >> 8, k = idx & 255;
        Wsh[j * 258 + k] = W_hh[idx];
    }
    // Stage h0 slice
    for (int idx = tid; idx < 16 * HH; idx += 512) {
        int r = idx >> 8, k = idx & 255;
        Hs[r * 258 + k] = h0[(size_t)(b0 + r) * HH + k];
    }
    __syncthreads();

    const int n0 = wave * 16;
    float vals[8];

    for (int l = 0; l < LSEQ; ++l) {
        v8f acc0 = {};
        v8f acc1 = {};
        #pragma unroll
        for (int k0 = 0; k0 < HH; k0 += 8) {
            v2f a0 = *reinterpret_cast<const v2f*>(&Hs[col * 258 + k0 + 2 * hi]);
            v2f b0 = *reinterpret_cast<const v2f*>(&Wsh[(n0 + col) * 258 + k0 + 2 * hi]);
            v2f a1 = *reinterpret_cast<const v2f*>(&Hs[col * 258 + k0 + 4 + 2 * hi]);
            v2f b1 = *reinterpret_cast<const v2f*>(&Wsh[(n0 + col) * 258 + k0 + 4 + 2 * hi]);
            acc0 = __builtin_amdgcn_wmma_f32_16x16x4_f32(
                false, a0, false, b0, (short)0, acc0, false, false);
            acc1 = __builtin_amdgcn_wmma_f32_16x16x4_f32(
                false, a1, false, b1, (short)0, acc1, false, false);
        }
        v8f acc = acc0 + acc1;

        const size_t base = ((size_t)l * BB + b0) * HH;
        #pragma unroll
        for (int r = 0; r < 8; ++r) {
            int row = r + 8 * hi;          // local batch row in slice
            size_t idx = base + (size_t)row * HH + n0 + col;
            float v = tanhf(acc[r] + out[idx]);  // out currently holds xW[l]
            vals[r] = v;
            out[idx] = v;                  // overwrite with h[l]
        }

        __syncthreads();                   // all k-loop reads of Hs done
        #pragma unroll
        for (int r = 0; r < 8; ++r)
            Hs[(r + 8 * hi) * 258 + (n0 + col)] = vals[r];
        __syncthreads();                   // h updated before next step
    }

    #pragma unroll
    for (int r = 0; r < 8; ++r)
        hlast[(size_t)(b0 + r + 8 * hi) * HH + n0 + col] = vals[r];
}

// ---------------------------------------------------------------------------
extern "C" void kernel_launch(void* const* d_in, const int* in_sizes, int n_in,
                              void* d_out, int out_size, void* d_ws, size_t ws_size,
                              hipStream_t stream) {
    (void)in_sizes; (void)n_in; (void)out_size; (void)d_ws; (void)ws_size;

    const float* x    = (const float*)d_in[0];
    const float* h0   = (const float*)d_in[1];
    const float* W_ih = (const float*)d_in[2];
    const float* W_hh = (const float*)d_in[3];
    const float* bias = (const float*)d_in[4];

    float* out   = (float*)d_out;                       // [L,B,H] outputs
    float* hlast = out + (size_t)LSEQ * BB * HH;        // [B,H] h_last tail

    // Phase 1: parallel input projection into the outputs region (as xW)
    rnn_xproj_wmma<<<(LSEQ * BB) / 16, 512, 0, stream>>>(x, W_ih, bias, out);

    // Phase 2: 4 independent batch-sliced scans, W_hh + h in LDS (280.7 KB)
    const size_t smem_bytes = (size_t)(256 * 258 + 16 * 258) * sizeof(float);
    hipFuncSetAttribute(reinterpret_cast<const void*>(rnn_scan_wmma),
                        hipFuncAttributeMaxDynamicSharedMemorySize,
                        (int)smem_bytes);
    rnn_scan_wmma<<<4, 512, smem_bytes, stream>>>(W_hh, h0, out, hlast);
}